// SparseDropout_52381421142407
// MI455X (gfx1250) — compile-verified
//
#include <hip/hip_runtime.h>
#include <hip/hip_bf16.h>

// Sparse dropout: out[i] = keep_mask[i] ? values[i] * (1/(1-p)) : 0
// Pure streaming op, 0.1 FLOP/byte -> HBM/L2 bandwidth bound on MI455X.
// 180 MB total traffic fits the 192 MB global L2, so use default (RT)
// cache hints and 128-bit vector memory ops in wave32.

#define DROPOUT_P 0.1f

typedef float v4f __attribute__((ext_vector_type(4)));

__global__ __launch_bounds__(256) void sparse_dropout_vec4_kernel(
    const float* __restrict__ values,
    const unsigned char* __restrict__ keep_mask,
    float* __restrict__ out,
    int nvec)  // number of 4-element chunks
{
    const float scale = 1.0f / (1.0f - DROPOUT_P);
    int i = blockIdx.x * blockDim.x + threadIdx.x;
    if (i >= nvec) return;

    const size_t base = (size_t)i * 4;

    // 128-bit load of 4 values (global_load_b128) and one dword of 4 mask
    // bytes (global_load_b32). JAX/numpy bool arrays are 1 byte per element.
    const v4f v = *reinterpret_cast<const v4f*>(values + base);
    const unsigned int m = *reinterpret_cast<const unsigned int*>(keep_mask + base);

    v4f r;
    r.x = (m & 0x000000ffu) ? v.x * scale : 0.0f;
    r.y = (m & 0x0000ff00u) ? v.y * scale : 0.0f;
    r.z = (m & 0x00ff0000u) ? v.z * scale : 0.0f;
    r.w = (m & 0xff000000u) ? v.w * scale : 0.0f;

    // 128-bit store (global_store_b128)
    *reinterpret_cast<v4f*>(out + base) = r;
}

// Scalar tail for n % 4 elements (NNZ = 20,000,000 is divisible by 4, but
// stay general/deterministic for any size).
__global__ __launch_bounds__(64) void sparse_dropout_tail_kernel(
    const float* __restrict__ values,
    const unsigned char* __restrict__ keep_mask,
    float* __restrict__ out,
    int start, int n)
{
    const float scale = 1.0f / (1.0f - DROPOUT_P);
    int i = start + blockIdx.x * blockDim.x + threadIdx.x;
    if (i >= n) return;
    out[i] = keep_mask[i] ? values[i] * scale : 0.0f;
}

extern "C" void kernel_launch(void* const* d_in, const int* in_sizes, int n_in,
                              void* d_out, int out_size, void* d_ws, size_t ws_size,
                              hipStream_t stream) {
    // setup_inputs order: indices (2*NNZ int32), values (NNZ f32), keep_mask (NNZ bool)
    const float*         values = (const float*)d_in[1];
    const unsigned char* mask   = (const unsigned char*)d_in[2];
    float*               out    = (float*)d_out;

    const int n    = in_sizes[1];   // NNZ
    const int nvec = n / 4;
    const int rem  = n - nvec * 4;

    if (nvec > 0) {
        dim3 block(256);
        dim3 grid((nvec + 255) / 256);
        sparse_dropout_vec4_kernel<<<grid, block, 0, stream>>>(values, mask, out, nvec);
    }
    if (rem > 0) {
        sparse_dropout_tail_kernel<<<1, 64, 0, stream>>>(values, mask, out, nvec * 4, n);
    }
}